// Graph_Learn_40965398069650
// MI455X (gfx1250) — compile-verified
//
#include <hip/hip_runtime.h>
#include <hip/hip_bf16.h>
#include <math.h>

typedef float v2f __attribute__((ext_vector_type(2)));
typedef float v8f __attribute__((ext_vector_type(8)));

constexpr int kN = 512;        // nodes per slab
constexpr int kF = 32;         // features
constexpr int kWaves = 4;      // waves per block (wave32)
constexpr int kEStride = 516;  // padded e-row stride (bank-conflict-free: 516 % 64 == 4)

// Per block: one (b,t) slab section, 64 rows (16 rows per wave).
// Staging: x slab -> LDS via gfx1250 async-to-LDS DMA (ASYNCcnt path, no VGPR round-trip).
// Pass 1: each lane computes e = exp(-sum_f w_f |x_i - x_j|) for its row i = lane&15,
//         2 j's per 4-wide j-step; feeds them as the B matrix of
//         V_WMMA_F32_16X16X4_F32 with A = ones -> accumulates per-row sums.
//         (A = ones makes the result invariant to the K-order inside the B layout;
//          we only rely on N = lane&15, per the ISA B/C/D layout.)
// Pass 2: S = e * (1/rowsum), coalesced stores.
__global__ __launch_bounds__(128) void
graph_softmax_kernel(const float* __restrict__ x,
                     const float* __restrict__ a,
                     float* __restrict__ out)
{
    extern __shared__ float smem[];
    float* xs = smem;                                  // [kN*kF] staged slab
    float* es = smem + kN * kF;                        // [kWaves*16*kEStride]
    float* rs = es + kWaves * 16 * kEStride;           // [kWaves*16] row sums

    const int tid  = threadIdx.x;
    const int lane = tid & 31;
    const int wv   = tid >> 5;

    const int blk  = blockIdx.x;
    const int slab = blk >> 3;   // b*T + t  (0..31)
    const int rb   = blk & 7;    // 64-row block within slab

    const float* xsrc = x + (size_t)slab * (kN * kF);
    float* osl = out + (size_t)slab * (size_t)(kN * kN);

    // ---- stage x slab into LDS via async DMA (GLOBAL_LOAD_ASYNC_TO_LDS_B128) ----
    // Flat addresses to LDS carry the wave-relative LDS offset in addr[31:0] (ISA 10.2),
    // which is exactly what the async instruction's vdst operand takes.
    {
        const unsigned lds_base = (unsigned)(uintptr_t)xs;
        const float4* src4 = (const float4*)xsrc;
        for (int idx = tid; idx < (kN * kF) / 4; idx += 128) {
            const unsigned lds_addr = lds_base + (unsigned)idx * 16u;
            const float4* gaddr = src4 + idx;
            asm volatile("global_load_async_to_lds_b128 %0, %1, off"
                         :: "v"(lds_addr), "v"(gaddr)
                         : "memory");
        }
        asm volatile("s_wait_asynccnt 0" ::: "memory");
    }
    __syncthreads();

    const int i_loc = lane & 15;
    const int half  = lane >> 4;          // selects which j inside a pair
    const int iBase = rb * 64 + wv * 16;
    const int i     = iBase + i_loc;

    // ---- preload x_i and w into registers (fully unrolled -> VGPRs) ----
    float xi[kF], wr[kF];
    #pragma unroll
    for (int f = 0; f < kF; ++f) {
        xi[f] = xs[i * kF + f];
        wr[f] = a[f];
    }

    float* erow = es + (wv * 16 + i_loc) * kEStride;

    v8f acc = {};                 // C/D accumulator: every element -> rowsum[lane&15]
    v2f ones;
    ones.x = 1.0f; ones.y = 1.0f; // A = 16x4 all-ones (layout-invariant)

    // ---- pass 1: e + wmma rowsum ----
    for (int jb = 0; jb < kN; jb += 4) {
        const int j0 = jb + half;       // B VGPR 0
        const int j1 = jb + 2 + half;   // B VGPR 1
        const float4* p0 = (const float4*)(xs + j0 * kF);
        const float4* p1 = (const float4*)(xs + j1 * kF);

        float d00 = 0.f, d01 = 0.f, d02 = 0.f, d03 = 0.f;
        float d10 = 0.f, d11 = 0.f, d12 = 0.f, d13 = 0.f;
        #pragma unroll
        for (int q = 0; q < kF / 4; ++q) {
            float4 u = p0[q];
            float4 v = p1[q];
            d00 += wr[4*q+0] * fabsf(xi[4*q+0] - u.x);
            d01 += wr[4*q+1] * fabsf(xi[4*q+1] - u.y);
            d02 += wr[4*q+2] * fabsf(xi[4*q+2] - u.z);
            d03 += wr[4*q+3] * fabsf(xi[4*q+3] - u.w);
            d10 += wr[4*q+0] * fabsf(xi[4*q+0] - v.x);
            d11 += wr[4*q+1] * fabsf(xi[4*q+1] - v.y);
            d12 += wr[4*q+2] * fabsf(xi[4*q+2] - v.z);
            d13 += wr[4*q+3] * fabsf(xi[4*q+3] - v.w);
        }
        const float e0 = __expf(-((d00 + d01) + (d02 + d03)));
        const float e1 = __expf(-((d10 + d11) + (d12 + d13)));
        erow[j0] = e0;
        erow[j1] = e1;

        v2f b; b.x = e0; b.y = e1;
        // D = Ones(16x4) x E(4x16) + C : accumulates column sums over K=4 j's,
        // i.e. partial rowsums of row i = N = lane&15, replicated over M.
        acc = __builtin_amdgcn_wmma_f32_16x16x4_f32(
            false, ones, false, b, (short)0, acc, false, false);
    }

    // every element of acc == rowsum[lane&15]; publish (duplicate writes are identical)
    rs[wv * 16 + i_loc] = acc[0];

    // ---- pass 2: normalize and store (wave-private data; in-wave LDS ordering) ----
    for (int r = 0; r < 16; ++r) {
        const float inv = 1.0f / rs[wv * 16 + r];
        const float* er = es + (wv * 16 + r) * kEStride;
        float* orow = osl + (size_t)(iBase + r) * kN;
        for (int c = lane; c < kN; c += 32) {
            orow[c] = er[c] * inv;
        }
    }
}

extern "C" void kernel_launch(void* const* d_in, const int* in_sizes, int n_in,
                              void* d_out, int out_size, void* d_ws, size_t ws_size,
                              hipStream_t stream) {
    (void)in_sizes; (void)n_in; (void)d_ws; (void)ws_size; (void)out_size;
    const float* x = (const float*)d_in[0];   // (4,8,512,32) f32
    const float* a = (const float*)d_in[1];   // (32,1) f32
    float* out = (float*)d_out;               // (4,8,512,512) f32

    const size_t shmem =
        (size_t)(kN * kF + kWaves * 16 * kEStride + kWaves * 16) * sizeof(float);

    // Allow >64KB dynamic LDS (gfx1250 WGP has 320KB); deterministic every call.
    (void)hipFuncSetAttribute((const void*)graph_softmax_kernel,
                              hipFuncAttributeMaxDynamicSharedMemorySize,
                              (int)shmem);

    dim3 grid(256);
    dim3 block(128);
    hipLaunchKernelGGL(graph_softmax_kernel, grid, block, shmem, stream, x, a, out);
}